// MaxGraphPool_14963666059305
// MI455X (gfx1250) — compile-verified
//
#include <hip/hip_runtime.h>
#include <math.h>
#include <stdint.h>

typedef __attribute__((ext_vector_type(2))) float v2f;
typedef __attribute__((ext_vector_type(8))) float v8f;
typedef __attribute__((ext_vector_type(4))) int   v4i;

#define DIN 128
#define DOUT 128
#define CHUNK 256      // nodes per workgroup
#define SUB 64         // nodes per LDS sub-tile
#define LDS_STRIDE 132 // 128 + 4 pad: kills 16-way bank conflicts on column reads

// ---- CDNA5 async global->LDS copy path (guarded; falls back to float4 copy) ----
#if defined(__HIP_DEVICE_COMPILE__)
#if __has_builtin(__builtin_amdgcn_global_load_async_to_lds_b128)
#define HAS_ASYNC_COPY 1
typedef __attribute__((address_space(1))) v4i Gv4i;  // global (prints as __device__)
typedef __attribute__((address_space(3))) v4i Lv4i;  // LDS
#endif
#endif

__device__ inline void waitAsync0() {
#if defined(__HIP_DEVICE_COMPILE__)
#if __has_builtin(__builtin_amdgcn_s_wait_asynccnt)
  __builtin_amdgcn_s_wait_asynccnt(0);
#else
  asm volatile("s_wait_asynccnt 0" ::: "memory");
#endif
#endif
}

__device__ inline void atomicMaxF32(float* addr, float val) {
  int old = __float_as_int(*addr);
  while (__int_as_float(old) < val) {
    int assumed = old;
    old = atomicCAS((int*)addr, assumed, __float_as_int(val));
    if (old == assumed) break;
  }
}

__global__ void mgp_init(float* __restrict__ pooled, int n) {
  int i = blockIdx.x * blockDim.x + threadIdx.x;
  if (i < n) pooled[i] = -INFINITY;
}

__global__ __launch_bounds__(256) void mgp_main(
    const float* __restrict__ x, const float* __restrict__ W,
    const float* __restrict__ bvec, float* __restrict__ pooled,
    int N, int chunksPerBatch) {
  extern __shared__ float smem[];
  float* xs = smem;                      // [SUB][LDS_STRIDE] x tile
  float* gs = smem + SUB * LDS_STRIDE;   // [SUB][LDS_STRIDE] gate tile

  const int tid  = threadIdx.x;
  const int lane = tid & 31;
  const int wave = tid >> 5;
  const int b     = blockIdx.x / chunksPerBatch;
  const int chunk = blockIdx.x % chunksPerBatch;

  // ---- preload this wave's B fragments (one 16-col N-tile of W), reused all kernel ----
  // WMMA f32 16x16x4 B layout: lane l holds (K = ks*4 + (l>>4)*2 + j, N = nt*16 + (l&15))
  const int nt   = wave;                 // 8 waves cover 128 output cols
  const int ocol = nt * 16 + (lane & 15);
  const int krow = (lane >> 4) * 2;
  v2f bf[32];
#pragma unroll
  for (int ks = 0; ks < 32; ++ks) {
    int k = ks * 4 + krow;
    bf[ks].x = W[k * DOUT + ocol];
    bf[ks].y = W[(k + 1) * DOUT + ocol];
  }
  const float bias = bvec[ocol];

  // ---- per-thread 8x8 (i,o) pooled accumulator block, packed as float2 ----
  const int i0 = (tid >> 4) * 8;
  const int o0 = (tid & 15) * 8;
  v2f acc2[8][4];
#pragma unroll
  for (int i = 0; i < 8; ++i)
#pragma unroll
    for (int j = 0; j < 4; ++j) acc2[i][j] = (v2f){-INFINITY, -INFINITY};

  const float* xg = x + ((size_t)b * N + (size_t)chunk * CHUNK) * DIN;

  for (int sub = 0; sub < CHUNK / SUB; ++sub) {
    // ---- stage SUB x DIN tile of x into padded LDS ----
    const float4* src = (const float4*)(xg + (size_t)sub * SUB * DIN);
#pragma unroll
    for (int it = 0; it < (SUB * DIN / 4) / 256; ++it) {
      int idx = it * 256 + tid;     // float4 index; 32 float4 per row
      int r   = idx >> 5;
      int c4  = idx & 31;
      float* dst = xs + r * LDS_STRIDE + c4 * 4;
#if defined(HAS_ASYNC_COPY)
      // CDNA5 async-DMA path: global -> LDS without a VGPR round trip
      __builtin_amdgcn_global_load_async_to_lds_b128((Gv4i*)(src + idx), (Lv4i*)dst, 0, 0);
#else
      *(float4*)dst = src[idx];
#endif
    }
#if defined(HAS_ASYNC_COPY)
    waitAsync0();
#endif
    __syncthreads();

    // ---- Phase A: gate = sigmoid(x @ W + b) via V_WMMA_F32_16X16X4_F32 ----
    // Two independent accumulator chains so the XDL pipe isn't RAW-serialized.
#pragma unroll 1
    for (int mt = 0; mt < SUB / 16; ++mt) {
      v8f c0 = { bias, bias, bias, bias, bias, bias, bias, bias };
      v8f c1 = { 0.f, 0.f, 0.f, 0.f, 0.f, 0.f, 0.f, 0.f };
      const int mrow = mt * 16 + (lane & 15);
      const float* xrow = xs + mrow * LDS_STRIDE;
#pragma unroll
      for (int ks = 0; ks < 32; ks += 2) {
        int k0 = ks * 4 + krow;
        int k1 = (ks + 1) * 4 + krow;
        v2f a0, a1;
        a0.x = xrow[k0]; a0.y = xrow[k0 + 1];
        a1.x = xrow[k1]; a1.y = xrow[k1 + 1];
        c0 = __builtin_amdgcn_wmma_f32_16x16x4_f32(
                 false, a0, false, bf[ks], (short)0, c0, false, false);
        c1 = __builtin_amdgcn_wmma_f32_16x16x4_f32(
                 false, a1, false, bf[ks + 1], (short)0, c1, false, false);
      }
      v8f c = c0 + c1;
      // C/D layout: VGPR j, lane l -> M = mt*16 + j + (l>>4)*8, N = ocol
      const int mbase = mt * 16 + (lane >> 4) * 8;
#pragma unroll
      for (int j = 0; j < 8; ++j) {
        float g = 1.0f / (1.0f + __expf(-c[j]));
        gs[(mbase + j) * LDS_STRIDE + ocol] = g;
      }
    }
    __syncthreads();

    // ---- Phase B: outer-product max pooling, 8x8 register block per thread ----
    // float2 products encourage V_PK_MUL_F32 (VOP3P) packing: 32 pk_mul + 64 max per node.
#pragma unroll 1
    for (int n = 0; n < SUB; ++n) {
      const float* xr = xs + n * LDS_STRIDE;
      const float* gr = gs + n * LDS_STRIDE;
      float4 xa = *(const float4*)(xr + i0);
      float4 xb = *(const float4*)(xr + i0 + 4);
      float4 ga = *(const float4*)(gr + o0);
      float4 gb = *(const float4*)(gr + o0 + 4);
      float xv[8] = {xa.x, xa.y, xa.z, xa.w, xb.x, xb.y, xb.z, xb.w};
      v2f gv[4] = {{ga.x, ga.y}, {ga.z, ga.w}, {gb.x, gb.y}, {gb.z, gb.w}};
#pragma unroll
      for (int i = 0; i < 8; ++i) {
        v2f xi = {xv[i], xv[i]};
#pragma unroll
        for (int j = 0; j < 4; ++j) {
          v2f p = xi * gv[j];           // packed multiply (V_PK_MUL_F32)
          acc2[i][j].x = fmaxf(acc2[i][j].x, p.x);
          acc2[i][j].y = fmaxf(acc2[i][j].y, p.y);
        }
      }
    }
    __syncthreads();
  }

  // ---- deterministic cross-workgroup merge (float max is order-independent) ----
  float* pb = pooled + (size_t)b * DIN * DOUT;
#pragma unroll
  for (int i = 0; i < 8; ++i)
#pragma unroll
    for (int j = 0; j < 4; ++j) {
      atomicMaxF32(&pb[(i0 + i) * DOUT + (o0 + 2 * j)],     acc2[i][j].x);
      atomicMaxF32(&pb[(i0 + i) * DOUT + (o0 + 2 * j + 1)], acc2[i][j].y);
    }
}

__global__ void mgp_mean(const float* __restrict__ pooled, float* __restrict__ out) {
  int b = blockIdx.x;
  int o = threadIdx.x;
  const float* pb = pooled + (size_t)b * DIN * DOUT;
  float s = 0.0f;
  for (int i = 0; i < DIN; ++i) s += pb[i * DOUT + o];
  out[b * DOUT + o] = s * (1.0f / DIN);
}

extern "C" void kernel_launch(void* const* d_in, const int* in_sizes, int n_in,
                              void* d_out, int out_size, void* d_ws, size_t ws_size,
                              hipStream_t stream) {
  const float* x    = (const float*)d_in[0];
  const float* W    = (const float*)d_in[1];
  const float* bvec = (const float*)d_in[2];
  float* out    = (float*)d_out;
  float* pooled = (float*)d_ws;   // B * DIN * DOUT floats (256 KB for B=4)

  const int B = out_size / DOUT;                 // 4
  const int N = in_sizes[0] / (DIN * B);         // 8192
  const int chunksPerBatch = N / CHUNK;          // 32
  const int poolN = B * DIN * DOUT;

  mgp_init<<<(poolN + 255) / 256, 256, 0, stream>>>(pooled, poolN);

  size_t shmem = (size_t)2 * SUB * LDS_STRIDE * sizeof(float);  // ~66 KB of the 320 KB WGP LDS
  mgp_main<<<B * chunksPerBatch, 256, shmem, stream>>>(x, W, bvec, pooled, N, chunksPerBatch);

  mgp_mean<<<B, DOUT, 0, stream>>>(pooled, out);
}